// GNNAttackPredictor_3865470566499
// MI455X (gfx1250) — compile-verified
//
#include <hip/hip_runtime.h>
#include <hip/hip_bf16.h>
#include <math.h>

#define NNODES 50000
#define NEDGES 800000
#define EP     (NEDGES + NNODES)   // edges + self loops
#define HC     128
#define NHEAD  2
#define CPH    64

typedef __attribute__((ext_vector_type(2))) float v2f;
typedef __attribute__((ext_vector_type(8))) float v8f;

// ---------------- atomic helpers (native CDNA5 float atomics) ----------------
__device__ __forceinline__ void atomAddF(float* p, float v) {
    __hip_atomic_fetch_add(p, v, __ATOMIC_RELAXED, __HIP_MEMORY_SCOPE_AGENT);
}
// float atomic-max via ordered-int trick; storage initialized to 0xFFFFFFFF
__device__ __forceinline__ void atomMaxF(float* p, float v) {
    if (v >= 0.0f) atomicMax((int*)p, __float_as_int(v));
    else           atomicMin((unsigned int*)p, __float_as_uint(v));
}

// ---------------- WMMA fp32 GEMM: C[M,128] = act(A[M,K] @ B[K,128] + bias) ---
// grid.x = M/16 (M multiple of 16), block = 256 threads = 8 waves.
// Wave w computes the 16x16 tile at cols [16w, 16w+16). K in {32,128}.
__global__ void __launch_bounds__(256)
gemm_wmma_f32(const float* __restrict__ A, const float* __restrict__ B,
              const float* __restrict__ bias, float* __restrict__ C,
              int K, int has_bias, int do_relu) {
    __shared__ float ldsA[16 * (HC + 4)];     // padded to kill bank conflicts
    const int r0   = blockIdx.x * 16;
    const int ldsk = K + 4;

    // cooperative stage of the 16xK A tile
    for (int t = threadIdx.x; t < 16 * K; t += 256) {
        int r = t / K, c = t - r * K;
        ldsA[r * ldsk + c] = A[(size_t)(r0 + r) * K + c];
    }
    __syncthreads();

    const int lane  = threadIdx.x & 31;
    const int wv    = threadIdx.x >> 5;        // 0..7 -> column tile
    const int mrow  = lane & 15;               // M index for A frag / N index for B,C
    const int khalf = (lane < 16) ? 0 : 2;     // K pairing per ISA 16x4 f32 layout
    const int col   = (wv << 4) + mrow;

    v8f acc = {};
    for (int kb = 0; kb < K; kb += 4) {
        v2f a, b;
        // A 16x4: VGPR0 = {K=kb | K=kb+2}, VGPR1 = {K=kb+1 | K=kb+3}
        a.x = ldsA[mrow * ldsk + kb + khalf];
        a.y = ldsA[mrow * ldsk + kb + khalf + 1];
        // B 4x16 (row-major [K,128]): same lane-half K pairing, col = lane&15
        b.x = B[(size_t)(kb + khalf) * HC + col];
        b.y = B[(size_t)(kb + khalf + 1) * HC + col];
        acc = __builtin_amdgcn_wmma_f32_16x16x4_f32(
            false, a, false, b, (short)0, acc, false, false);
    }

    // C/D layout: VGPR r -> M = r + (lane<16 ? 0 : 8), N = lane&15
    const int rbase = r0 + ((lane < 16) ? 0 : 8);
    float bb = has_bias ? bias[col] : 0.0f;
    #pragma unroll
    for (int r = 0; r < 8; r++) {
        float v = acc[r] + bb;
        if (do_relu) v = fmaxf(v, 0.0f);
        C[(size_t)(rbase + r) * HC + col] = v;
    }
}

// ---------------- attention scalars: a_src/a_dst[n,h] = <h_lin[n,h,:], att> ---
__global__ void att_scalars(const float* __restrict__ hl,
                            const float* __restrict__ att_s,
                            const float* __restrict__ att_d,
                            float* __restrict__ a_src, float* __restrict__ a_dst) {
    int i = blockIdx.x * blockDim.x + threadIdx.x;   // (n,h) pairs
    if (i >= NNODES * NHEAD) return;
    int n = i >> 1, h = i & 1;
    const float* hp = hl + (size_t)n * HC + h * CPH;
    const float* ws = att_s + h * CPH;
    const float* wd = att_d + h * CPH;
    float s = 0.f, d = 0.f;
    for (int c = 0; c < CPH; c++) { float v = hp[c]; s += v * ws[c]; d += v * wd[c]; }
    a_src[i] = s; a_dst[i] = d;
}

// ---------------- edge pass 1: alpha = lrelu(a_src[s]+a_dst[d]); segmax ------
__global__ void edge_alpha_max(const int* __restrict__ ei,
                               const float* __restrict__ a_src,
                               const float* __restrict__ a_dst,
                               float* __restrict__ alpha, float* __restrict__ amax) {
    int e = blockIdx.x * blockDim.x + threadIdx.x;
    if (e >= EP) return;
    int s, d;
    if (e < NEDGES) { s = ei[e]; d = ei[NEDGES + e]; } else { s = d = e - NEDGES; }
    #pragma unroll
    for (int h = 0; h < NHEAD; h++) {
        float a = a_src[s * 2 + h] + a_dst[d * 2 + h];
        a = (a > 0.0f) ? a : a * 0.2f;                 // leaky_relu 0.2
        alpha[(size_t)e * 2 + h] = a;
        atomMaxF(&amax[d * 2 + h], a);
    }
}

// ---------------- edge pass 2: ex = exp(alpha - amax[d]); segsum denom -------
__global__ void edge_exp_sum(const int* __restrict__ ei,
                             float* __restrict__ alpha,
                             const float* __restrict__ amax,
                             float* __restrict__ denom) {
    int e = blockIdx.x * blockDim.x + threadIdx.x;
    if (e >= EP) return;
    int d;
    if (e < NEDGES) { d = ei[NEDGES + e]; } else { d = e - NEDGES; }
    #pragma unroll
    for (int h = 0; h < NHEAD; h++) {
        float ex = expf(alpha[(size_t)e * 2 + h] - amax[d * 2 + h]);
        alpha[(size_t)e * 2 + h] = ex;
        atomAddF(&denom[d * 2 + h], ex);
    }
}

// ---------------- edge pass 3: out[d] += h_lin[s] * (ex/denom[d]) ------------
// one wave per edge; lane handles 4 channels (float4 gather, 4 atomic adds)
__global__ void edge_scatter(const int* __restrict__ ei,
                             const float* __restrict__ exb,
                             const float* __restrict__ denom,
                             const float* __restrict__ hl,
                             float* __restrict__ out) {
    long long gid = (long long)blockIdx.x * blockDim.x + threadIdx.x;
    int e = (int)(gid >> 5);
    if (e >= EP) return;
    int lane = (int)(gid & 31);
    int s, d;
    if (e < NEDGES) { s = ei[e]; d = ei[NEDGES + e]; } else { s = d = e - NEDGES; }
    int h = lane >> 4;                 // lanes 0-15 -> head 0, 16-31 -> head 1
    int c = lane * 4;                  // channel base
    float coef = exb[(size_t)e * 2 + h] / denom[d * 2 + h];
    const float4 v = *(const float4*)(hl + (size_t)s * HC + c);
    float* o = out + (size_t)d * HC + c;
    atomAddF(o + 0, v.x * coef);
    atomAddF(o + 1, v.y * coef);
    atomAddF(o + 2, v.z * coef);
    atomAddF(o + 3, v.w * coef);
}

// ---------------- h = relu(h + bias) in place --------------------------------
__global__ void bias_relu(float* __restrict__ h, const float* __restrict__ bias) {
    int i = blockIdx.x * blockDim.x + threadIdx.x;
    if (i >= NNODES * HC) return;
    h[i] = fmaxf(h[i] + bias[i & (HC - 1)], 0.0f);
}

// ---------------- column mean partials (atomic into g[128]) ------------------
__global__ void col_sum(const float* __restrict__ h, float* __restrict__ g) {
    int c = threadIdx.x;                    // 128 threads
    float s = 0.f;
    for (int r = blockIdx.x; r < NNODES; r += gridDim.x)
        s += h[(size_t)r * HC + c];
    atomAddF(&g[c], s);
}

// ---------------- path head (single block) + esc output ----------------------
__global__ void head_path(const float* __restrict__ hA, const float* __restrict__ g,
                          const float* __restrict__ wp1, const float* __restrict__ bp1,
                          const float* __restrict__ wp2, const float* __restrict__ bp2,
                          const float* __restrict__ wp3, const float* __restrict__ bp3,
                          float* __restrict__ out) {
    __shared__ float pcat[256];
    __shared__ float p1[128];
    __shared__ float p2[64];
    int t = threadIdx.x;
    pcat[t] = (t < 128) ? hA[t] : g[t - 128] * (1.0f / (float)NNODES);
    __syncthreads();
    if (t < 128) {
        float s = bp1[t];
        for (int i = 0; i < 256; i++) s += pcat[i] * wp1[i * 128 + t];
        p1[t] = fmaxf(s, 0.0f);
    }
    __syncthreads();
    if (t < 64) {
        float s = bp2[t];
        for (int i = 0; i < 128; i++) s += p1[i] * wp2[i * 64 + t];
        p2[t] = fmaxf(s, 0.0f);
    }
    __syncthreads();
    if (t == 0) {
        float s = bp3[0];
        for (int i = 0; i < 64; i++) s += p2[i] * wp3[i];
        out[0] = 1.0f / (1.0f + expf(-s));      // path
        out[1 + NNODES] = 0.0f;                 // esc
    }
}

// ---------------- vulnerability head: thread per node ------------------------
__global__ void head_vuln(const float* __restrict__ hA,
                          const float* __restrict__ wv1, const float* __restrict__ bv1,
                          const float* __restrict__ wv2, const float* __restrict__ bv2,
                          float* __restrict__ out) {
    int n = blockIdx.x * blockDim.x + threadIdx.x;
    if (n >= NNODES) return;
    const float* hr = hA + (size_t)n * HC;
    float s2 = bv2[0];
    for (int j = 0; j < 64; j++) {
        float a = bv1[j];
        for (int i = 0; i < HC; i++) a += hr[i] * wv1[i * 64 + j];
        a = fmaxf(a, 0.0f);
        s2 += a * wv2[j];
    }
    out[n] = 1.0f / (1.0f + expf(-s2));
}

extern "C" void kernel_launch(void* const* d_in, const int* in_sizes, int n_in,
                              void* d_out, int out_size, void* d_ws, size_t ws_size,
                              hipStream_t stream) {
    (void)in_sizes; (void)n_in; (void)out_size; (void)ws_size;

    const float* x     = (const float*)d_in[0];
    const int*   ei    = (const int*)  d_in[1];
    // d_in[2] edge_attr, d_in[7] w_e, d_in[8] b_e: output unused by reference graph
    const float* w_n1  = (const float*)d_in[3];
    const float* b_n1  = (const float*)d_in[4];
    const float* w_n2  = (const float*)d_in[5];
    const float* b_n2  = (const float*)d_in[6];
    const float* convw = (const float*)d_in[9];
    const float* att_s = (const float*)d_in[10];
    const float* att_d = (const float*)d_in[11];
    const float* convb = (const float*)d_in[12];
    const float* wp1 = (const float*)d_in[13]; const float* bp1 = (const float*)d_in[14];
    const float* wp2 = (const float*)d_in[15]; const float* bp2 = (const float*)d_in[16];
    const float* wp3 = (const float*)d_in[17]; const float* bp3 = (const float*)d_in[18];
    const float* wv1 = (const float*)d_in[19]; const float* bv1 = (const float*)d_in[20];
    const float* wv2 = (const float*)d_in[21]; const float* bv2 = (const float*)d_in[22];
    float* out = (float*)d_out;

    // workspace carve-up (floats); ~60 MB total, L2-resident on MI455X
    float* ws    = (float*)d_ws;
    float* hA    = ws;                                  // [N,128] current features
    float* hB    = hA + (size_t)NNODES * HC;            // [N,128] linear proj
    float* a_src = hB + (size_t)NNODES * HC;            // [N,2]
    float* a_dst = a_src + NNODES * NHEAD;
    float* amax  = a_dst + NNODES * NHEAD;
    float* denom = amax  + NNODES * NHEAD;
    float* alpha = denom + NNODES * NHEAD;              // [EP,2]
    float* g     = alpha + (size_t)EP * NHEAD;          // [128]

    const int nb16 = NNODES / 16;                       // 3125, exact

    // node encoder: hB = relu(x@w1 + b1); hA = hB@w2 + b2
    gemm_wmma_f32<<<nb16, 256, 0, stream>>>(x,  w_n1, b_n1, hB, 32,  1, 1);
    gemm_wmma_f32<<<nb16, 256, 0, stream>>>(hB, w_n2, b_n2, hA, 128, 1, 0);

    for (int l = 0; l < 3; l++) {
        // h_lin = hA @ W_l  (no bias in GAT projection)
        gemm_wmma_f32<<<nb16, 256, 0, stream>>>(hA, convw + (size_t)l * HC * HC,
                                                nullptr, hB, 128, 0, 0);
        att_scalars<<<(NNODES * NHEAD + 255) / 256, 256, 0, stream>>>(
            hB, att_s + l * NHEAD * CPH, att_d + l * NHEAD * CPH, a_src, a_dst);

        hipMemsetAsync(amax,  0xFF, (size_t)NNODES * NHEAD * 4, stream); // -NaN init for max-trick
        hipMemsetAsync(denom, 0,    (size_t)NNODES * NHEAD * 4, stream);
        hipMemsetAsync(hA,    0,    (size_t)NNODES * HC * 4,    stream); // aggregate target

        edge_alpha_max<<<(EP + 255) / 256, 256, 0, stream>>>(ei, a_src, a_dst, alpha, amax);
        edge_exp_sum  <<<(EP + 255) / 256, 256, 0, stream>>>(ei, alpha, amax, denom);
        edge_scatter  <<<(int)(((long long)EP * 32 + 255) / 256), 256, 0, stream>>>(
            ei, alpha, denom, hB, hA);
        bias_relu<<<(NNODES * HC + 255) / 256, 256, 0, stream>>>(hA, convb + l * HC);
    }

    // graph mean + heads
    hipMemsetAsync(g, 0, 128 * 4, stream);
    col_sum  <<<256, 128, 0, stream>>>(hA, g);
    head_path<<<1, 256, 0, stream>>>(hA, g, wp1, bp1, wp2, bp2, wp3, bp3, out);
    head_vuln<<<(NNODES + 255) / 256, 256, 0, stream>>>(hA, wv1, bv1, wv2, bv2, out + 1);
}